// EdgeGatedGCN_50027779064050
// MI455X (gfx1250) — compile-verified
//
#include <hip/hip_runtime.h>
#include <hip/hip_bf16.h>
#include <math.h>

typedef float v2f __attribute__((ext_vector_type(2)));
typedef float v8f __attribute__((ext_vector_type(8)));

#define DF 64
#define STATS_BLOCKS 512

// ---------- helpers ----------
__device__ __forceinline__ unsigned fkey(float f) {
  unsigned u = __float_as_uint(f);
  return (u & 0x80000000u) ? ~u : (u | 0x80000000u);
}
__device__ __forceinline__ float funkey(unsigned u) {
  return (u & 0x80000000u) ? __uint_as_float(u & 0x7fffffffu)
                           : __uint_as_float(~u);
}
__device__ __forceinline__ float silu_f(float x) { return x / (1.0f + expf(-x)); }

// Wave-level fp32 WMMA GEMM: 16 rows x 64 cols, K=64.
// Arow: per-lane pointer to A row (lane&15). W row-major [64 out][64 in].
// acc[nt] = 16x16 tile for cols nt*16..nt*16+15.
__device__ __forceinline__ void wmma_k64(const float* __restrict__ Arow,
                                         const float* __restrict__ W,
                                         int kh, int lm, v8f acc[4]) {
  v2f areg[16];
#pragma unroll
  for (int kk = 0; kk < 16; ++kk) {
    int kb = kk * 4 + kh * 2;
    areg[kk].x = Arow[kb];
    areg[kk].y = Arow[kb + 1];
  }
#pragma unroll
  for (int nt = 0; nt < 4; ++nt) {
    const float* wrow = W + (long)(nt * 16 + lm) * DF;
    v8f c = {0.f, 0.f, 0.f, 0.f, 0.f, 0.f, 0.f, 0.f};
#pragma unroll
    for (int kk = 0; kk < 16; ++kk) {
      int kb = kk * 4 + kh * 2;
      v2f b;
      b.x = wrow[kb];
      b.y = wrow[kb + 1];
      c = __builtin_amdgcn_wmma_f32_16x16x4_f32(false, areg[kk], false, b,
                                                (short)0, c, false, false);
    }
    acc[nt] = c;
  }
}

// ---------- kernels ----------

// zero den/agg, set mx keys to 0 (== -inf under fkey ordering)
__global__ void k_init(unsigned* mx, float* den, float* agg, long total) {
  long i = (long)blockIdx.x * blockDim.x + threadIdx.x;
  long stride = (long)gridDim.x * blockDim.x;
  for (; i < total; i += stride) {
    mx[i] = 0u;
    den[i] = 0.0f;
    agg[i] = 0.0f;
  }
}

// e_lin[E,64] = concat(h[src], h[dst], e) @ W_upd^T + b_upd   (K = 192)
__global__ void k_edge_lin(const float* __restrict__ h, const float* __restrict__ e,
                           const int* __restrict__ src, const int* __restrict__ dst,
                           const float* __restrict__ Wupd, const float* __restrict__ bupd,
                           float* __restrict__ eLin, int E) {
  const int lane = threadIdx.x & 31;
  const int wave = threadIdx.x >> 5;
  const int lm = lane & 15;
  const int kh = lane >> 4;
  long tile = (long)blockIdx.x * (blockDim.x >> 5) + wave;
  long base = tile * 16;
  if (base >= (long)E) return;            // wave-uniform exit (EXEC stays full)
  long row = base + lm;
  if (row >= (long)E) row = (long)E - 1;  // clamp keeps EXEC all-ones
  const float* aS = h + (long)src[row] * DF;
  const float* aD = h + (long)dst[row] * DF;
  const float* aE = e + row * DF;
#pragma unroll
  for (int nt = 0; nt < 4; ++nt) {
    const int n = nt * 16 + lm;
    const float* wrow = Wupd + (long)n * 192;
    v8f c = {0.f, 0.f, 0.f, 0.f, 0.f, 0.f, 0.f, 0.f};
#pragma unroll
    for (int kk = 0; kk < 48; ++kk) {
      int kb = kk * 4 + kh * 2;
      int k1 = kb + 1;
      v2f a, b;
      a.x = (kb < 64) ? aS[kb] : (kb < 128) ? aD[kb - 64] : aE[kb - 128];
      a.y = (k1 < 64) ? aS[k1] : (k1 < 128) ? aD[k1 - 64] : aE[k1 - 128];
      b.x = wrow[kb];
      b.y = wrow[k1];
      c = __builtin_amdgcn_wmma_f32_16x16x4_f32(false, a, false, b,
                                                (short)0, c, false, false);
    }
    const float bias = bupd[n];
#pragma unroll
    for (int v = 0; v < 8; ++v) {
      long r = base + v + kh * 8;
      if (r < (long)E) eLin[r * DF + n] = c[v] + bias;
    }
  }
}

// deterministic per-feature sum/sumsq partials over `rows` rows
__global__ void k_colstats(const float* __restrict__ X, long rows,
                           float* __restrict__ partial) {
  __shared__ float sS[4][DF];
  __shared__ float sQ[4][DF];
  const int f = threadIdx.x & 63;
  const int g = threadIdx.x >> 6;  // 0..3
  float s = 0.0f, q = 0.0f;
  for (long r = (long)blockIdx.x * 4 + g; r < rows; r += (long)gridDim.x * 4) {
    float v = X[r * DF + f];
    s += v;
    q += v * v;
  }
  sS[g][f] = s;
  sQ[g][f] = q;
  __syncthreads();
  if (g == 0) {
    float ts = sS[0][f] + sS[1][f] + sS[2][f] + sS[3][f];
    float tq = sQ[0][f] + sQ[1][f] + sQ[2][f] + sQ[3][f];
    partial[(long)blockIdx.x * 128 + f] = ts;
    partial[(long)blockIdx.x * 128 + 64 + f] = tq;
  }
}

__global__ void k_finalize_stats(const float* __restrict__ partial, int nblocks,
                                 float rowsInv, float* __restrict__ stats) {
  int f = threadIdx.x;
  if (f < DF) {
    float s = 0.0f, q = 0.0f;
    for (int b = 0; b < nblocks; ++b) {
      s += partial[(long)b * 128 + f];
      q += partial[(long)b * 128 + 64 + f];
    }
    float m = s * rowsInv;
    float var = q * rowsInv - m * m;
    stats[f] = m;
    stats[64 + f] = rsqrtf(var + 1e-5f);
  }
}

// e2 = e + silu(batchnorm(e_lin))
__global__ void k_edge_bn(const float* __restrict__ e, const float* __restrict__ eLin,
                          const float* __restrict__ stats, const float* __restrict__ gamma,
                          const float* __restrict__ beta, float* __restrict__ e2,
                          long total) {
  long i = (long)blockIdx.x * blockDim.x + threadIdx.x;
  long stride = (long)gridDim.x * blockDim.x;
  for (; i < total; i += stride) {
    int f = (int)(i & 63);
    float x = (eLin[i] - stats[f]) * stats[64 + f] * gamma[f] + beta[f];
    e2[i] = e[i] + silu_f(x);
  }
}

// logit = e2 @ W_act^T + b_act ; segment-max over dst via monotone-uint atomicMax
__global__ void k_logit_max(const float* __restrict__ e2, const int* __restrict__ dst,
                            const float* __restrict__ Wact, const float* __restrict__ bact,
                            float* __restrict__ logit, unsigned* __restrict__ mx, int E) {
  const int lane = threadIdx.x & 31;
  const int wave = threadIdx.x >> 5;
  const int lm = lane & 15;
  const int kh = lane >> 4;
  long tile = (long)blockIdx.x * (blockDim.x >> 5) + wave;
  long base = tile * 16;
  if (base >= (long)E) return;
  long row = base + lm;
  if (row >= (long)E) row = (long)E - 1;
  v8f acc[4];
  wmma_k64(e2 + row * DF, Wact, kh, lm, acc);
  float bias[4];
#pragma unroll
  for (int nt = 0; nt < 4; ++nt) bias[nt] = bact[nt * 16 + lm];
#pragma unroll
  for (int v = 0; v < 8; ++v) {
    long r = base + v + kh * 8;
    if (r < (long)E) {
      int d = dst[r];
#pragma unroll
      for (int nt = 0; nt < 4; ++nt) {
        float val = acc[nt][v] + bias[nt];
        logit[r * DF + nt * 16 + lm] = val;
        atomicMax(&mx[(long)d * DF + nt * 16 + lm], fkey(val));
      }
    }
  }
}

// den[dst] += exp(logit - mx[dst])
__global__ void k_exp_den(const float* __restrict__ logit, const int* __restrict__ dst,
                          const unsigned* __restrict__ mx, float* __restrict__ den,
                          long total) {
  long i = (long)blockIdx.x * blockDim.x + threadIdx.x;
  long stride = (long)gridDim.x * blockDim.x;
  for (; i < total; i += stride) {
    long r = i >> 6;
    int f = (int)(i & 63);
    long doff = (long)dst[r] * DF + f;
    float ex = expf(logit[i] - funkey(mx[doff]));
    atomicAdd(&den[doff], ex);
  }
}

// hgd = h @ W_gd^T + b_gd ; hgs = h @ W_gs^T + b_gs
__global__ void k_node_lin(const float* __restrict__ h,
                           const float* __restrict__ Wgd, const float* __restrict__ bgd,
                           const float* __restrict__ Wgs, const float* __restrict__ bgs,
                           float* __restrict__ hgd, float* __restrict__ hgs, int N) {
  const int lane = threadIdx.x & 31;
  const int wave = threadIdx.x >> 5;
  const int lm = lane & 15;
  const int kh = lane >> 4;
  long tile = (long)blockIdx.x * (blockDim.x >> 5) + wave;
  long base = tile * 16;
  if (base >= (long)N) return;
  long row = base + lm;
  if (row >= (long)N) row = (long)N - 1;
  const float* Arow = h + row * DF;
  v8f acc[4];
  wmma_k64(Arow, Wgd, kh, lm, acc);
#pragma unroll
  for (int nt = 0; nt < 4; ++nt) {
    float bias = bgd[nt * 16 + lm];
#pragma unroll
    for (int v = 0; v < 8; ++v) {
      long r = base + v + kh * 8;
      if (r < (long)N) hgd[r * DF + nt * 16 + lm] = acc[nt][v] + bias;
    }
  }
  wmma_k64(Arow, Wgs, kh, lm, acc);
#pragma unroll
  for (int nt = 0; nt < 4; ++nt) {
    float bias = bgs[nt * 16 + lm];
#pragma unroll
    for (int v = 0; v < 8; ++v) {
      long r = base + v + kh * 8;
      if (r < (long)N) hgs[r * DF + nt * 16 + lm] = acc[nt][v] + bias;
    }
  }
}

// agg[dst] += (exp(logit-mx[dst])/den[dst]) * hgd[dst]
__global__ void k_aggregate(const float* __restrict__ logit, const int* __restrict__ dst,
                            const unsigned* __restrict__ mx, const float* __restrict__ den,
                            const float* __restrict__ hgd, float* __restrict__ agg,
                            long total) {
  long i = (long)blockIdx.x * blockDim.x + threadIdx.x;
  long stride = (long)gridDim.x * blockDim.x;
  for (; i < total; i += stride) {
    long r = i >> 6;
    int f = (int)(i & 63);
    long doff = (long)dst[r] * DF + f;
    float ex = expf(logit[i] - funkey(mx[doff]));
    float a = ex / den[doff];
    atomicAdd(&agg[doff], a * hgd[doff]);
  }
}

// hgs += agg  (h2a, in place)
__global__ void k_add_agg(float* __restrict__ hgs, const float* __restrict__ agg,
                          long total) {
  long i = (long)blockIdx.x * blockDim.x + threadIdx.x;
  long stride = (long)gridDim.x * blockDim.x;
  for (; i < total; i += stride) hgs[i] += agg[i];
}

// h2pre = h + silu(batchnorm(h2a))
__global__ void k_node_bn(const float* __restrict__ h, const float* __restrict__ h2a,
                          const float* __restrict__ stats, const float* __restrict__ gamma,
                          const float* __restrict__ beta, float* __restrict__ out,
                          long total) {
  long i = (long)blockIdx.x * blockDim.x + threadIdx.x;
  long stride = (long)gridDim.x * blockDim.x;
  for (; i < total; i += stride) {
    int f = (int)(i & 63);
    float x = (h2a[i] - stats[f]) * stats[64 + f] * gamma[f] + beta[f];
    out[i] = h[i] + silu_f(x);
  }
}

// h2 = h2pre @ W_out^T + b_out
__global__ void k_out(const float* __restrict__ h2pre, const float* __restrict__ Wout,
                      const float* __restrict__ bout, float* __restrict__ out, int N) {
  const int lane = threadIdx.x & 31;
  const int wave = threadIdx.x >> 5;
  const int lm = lane & 15;
  const int kh = lane >> 4;
  long tile = (long)blockIdx.x * (blockDim.x >> 5) + wave;
  long base = tile * 16;
  if (base >= (long)N) return;
  long row = base + lm;
  if (row >= (long)N) row = (long)N - 1;
  v8f acc[4];
  wmma_k64(h2pre + row * DF, Wout, kh, lm, acc);
#pragma unroll
  for (int nt = 0; nt < 4; ++nt) {
    float bias = bout[nt * 16 + lm];
#pragma unroll
    for (int v = 0; v < 8; ++v) {
      long r = base + v + kh * 8;
      if (r < (long)N) out[r * DF + nt * 16 + lm] = acc[nt][v] + bias;
    }
  }
}

// ---------- launcher ----------
extern "C" void kernel_launch(void* const* d_in, const int* in_sizes, int n_in,
                              void* d_out, int out_size, void* d_ws, size_t ws_size,
                              hipStream_t stream) {
  (void)n_in; (void)out_size; (void)ws_size;
  const float* h     = (const float*)d_in[0];
  const float* e     = (const float*)d_in[1];
  const int*   src   = (const int*)d_in[2];
  const int*   dst   = (const int*)d_in[3];
  const float* W_upd = (const float*)d_in[4];
  const float* b_upd = (const float*)d_in[5];
  const float* g_un  = (const float*)d_in[6];
  const float* be_un = (const float*)d_in[7];
  const float* W_act = (const float*)d_in[8];
  const float* b_act = (const float*)d_in[9];
  const float* W_gd  = (const float*)d_in[10];
  const float* b_gd  = (const float*)d_in[11];
  const float* W_gs  = (const float*)d_in[12];
  const float* b_gs  = (const float*)d_in[13];
  const float* g_gn  = (const float*)d_in[14];
  const float* be_gn = (const float*)d_in[15];
  const float* W_out = (const float*)d_in[16];
  const float* b_out = (const float*)d_in[17];

  const int N = in_sizes[0] / DF;
  const int E = in_sizes[2];
  const long NN = (long)N * DF;
  const long NE = (long)E * DF;

  float*    ws      = (float*)d_ws;
  float*    eLin    = ws;                       // E*64, reused as logit
  unsigned* mx      = (unsigned*)(ws + NE);     // N*64
  float*    den     = ws + NE + NN;             // N*64
  float*    agg     = den + NN;                 // N*64
  float*    hgd     = agg + NN;                 // N*64, reused as h2pre
  float*    hgs     = hgd + NN;                 // N*64, reused as h2a
  float*    partial = hgs + NN;                 // STATS_BLOCKS*128
  float*    statsE  = partial + (long)STATS_BLOCKS * 128;  // 128
  float*    statsN  = statsE + 128;                        // 128

  float* outH = (float*)d_out;   // h2 [N,64]
  float* outE = outH + NN;       // e2 [E,64]

  const int edgeTiles = (E + 15) / 16;
  const int nodeTiles = (N + 15) / 16;
  const int edgeBlocks = (edgeTiles + 3) / 4;   // 4 waves / block
  const int nodeBlocks = (nodeTiles + 3) / 4;
  const int blkNN = (int)((NN + 255) / 256);
  const int blkNE = (int)((NE + 255) / 256);

  k_init<<<blkNN, 256, 0, stream>>>(mx, den, agg, NN);
  k_edge_lin<<<edgeBlocks, 128, 0, stream>>>(h, e, src, dst, W_upd, b_upd, eLin, E);
  k_colstats<<<STATS_BLOCKS, 256, 0, stream>>>(eLin, (long)E, partial);
  k_finalize_stats<<<1, 64, 0, stream>>>(partial, STATS_BLOCKS, 1.0f / (float)E, statsE);
  k_edge_bn<<<blkNE, 256, 0, stream>>>(e, eLin, statsE, g_un, be_un, outE, NE);
  k_logit_max<<<edgeBlocks, 128, 0, stream>>>(outE, dst, W_act, b_act, eLin, mx, E);
  k_exp_den<<<blkNE, 256, 0, stream>>>(eLin, dst, mx, den, NE);
  k_node_lin<<<nodeBlocks, 128, 0, stream>>>(h, W_gd, b_gd, W_gs, b_gs, hgd, hgs, N);
  k_aggregate<<<blkNE, 256, 0, stream>>>(eLin, dst, mx, den, hgd, agg, NE);
  k_add_agg<<<blkNN, 256, 0, stream>>>(hgs, agg, NN);
  k_colstats<<<STATS_BLOCKS, 256, 0, stream>>>(hgs, (long)N, partial);
  k_finalize_stats<<<1, 64, 0, stream>>>(partial, STATS_BLOCKS, 1.0f / (float)N, statsN);
  k_node_bn<<<blkNN, 256, 0, stream>>>(h, hgs, statsN, g_gn, be_gn, hgd, NN);
  k_out<<<nodeBlocks, 128, 0, stream>>>(hgd, W_out, b_out, outH, N);
}